// DKD_19250043421298
// MI455X (gfx1250) — compile-verified
//
#include <hip/hip_runtime.h>
#include <math.h>

// Problem constants (match reference)
#define Bn   2
#define Hh   512
#define Ww   512
#define HW   (Hh*Ww)
#define Cc   128
#define Kk   2048
#define NMSR 3
#define RAD  2

typedef __attribute__((ext_vector_type(2))) float v2f;
typedef __attribute__((ext_vector_type(8))) float v8f;

__device__ __forceinline__ int clampi(int v, int lo, int hi){ return v < lo ? lo : (v > hi ? hi : v); }

// ---------------- NMS: separable 7-wide max-pool (fused variants) ----------------
__global__ void k_mp_rows(const float* __restrict__ in, float* __restrict__ out){
    int i = blockIdx.x * blockDim.x + threadIdx.x;
    if (i >= Bn*HW) return;
    int x = i % Ww; int base = i - x;
    float m = -INFINITY;
    #pragma unroll
    for (int dx = -NMSR; dx <= NMSR; ++dx){
        int xx = x + dx;
        if (xx >= 0 && xx < Ww) m = fmaxf(m, in[base + xx]);
    }
    out[i] = m;
}

__device__ __forceinline__ float colmax_split(const float* __restrict__ in, int b, int y, int x){
    int yLo = (y < Hh/2) ? 0 : Hh/2;       // max-pool does not cross the half split
    int yHi = yLo + Hh/2 - 1;
    float m = -INFINITY;
    #pragma unroll
    for (int dy = -NMSR; dy <= NMSR; ++dy){
        int yy = y + dy;
        if (yy >= yLo && yy <= yHi) m = fmaxf(m, in[(b*Hh + yy)*Ww + x]);
    }
    return m;
}

// pooled = colmax(rowpooled); mm = (s == pooled)
__global__ void k_cols_eq(const float* __restrict__ rowp, const float* __restrict__ s,
                          float* __restrict__ mm){
    int i = blockIdx.x * blockDim.x + threadIdx.x;
    if (i >= Bn*HW) return;
    int x = i % Ww; int y = (i / Ww) % Hh; int b = i / HW;
    float pooled = colmax_split(rowp, b, y, x);
    mm[i] = (s[i] == pooled) ? 1.0f : 0.0f;
}

// mmP = colmax(rowpooled mm); ss = (mmP>0) ? 0 : s
__global__ void k_cols_suppress(const float* __restrict__ rowp, const float* __restrict__ s,
                                float* __restrict__ mmP, float* __restrict__ ss){
    int i = blockIdx.x * blockDim.x + threadIdx.x;
    if (i >= Bn*HW) return;
    int x = i % Ww; int y = (i / Ww) % Hh; int b = i / HW;
    float p = colmax_split(rowp, b, y, x);
    mmP[i] = p;
    ss[i]  = (p > 0.0f) ? 0.0f : s[i];
}

// ssP = colmax(rowpooled ss); mm |= (ss==ssP) & ~(mmP>0)
__global__ void k_cols_combine(const float* __restrict__ rowp, const float* __restrict__ ss,
                               const float* __restrict__ mmP, float* __restrict__ mm){
    int i = blockIdx.x * blockDim.x + threadIdx.x;
    if (i >= Bn*HW) return;
    int x = i % Ww; int y = (i / Ww) % Hh; int b = i / HW;
    float ssP = colmax_split(rowp, b, y, x);
    bool sm = mmP[i] > 0.0f;
    bool nm = (ss[i] == ssP);
    bool m  = (mm[i] != 0.0f) || (nm && !sm);
    mm[i] = m ? 1.0f : 0.0f;
}

__global__ void k_nms_border(const float* __restrict__ s, const float* __restrict__ mm, float* __restrict__ nms){
    int i = blockIdx.x * blockDim.x + threadIdx.x;
    if (i >= Bn*HW) return;
    int x = i % Ww; int y = (i / Ww) % Hh;
    bool border = (y >= RAD+1) && (y < Hh-RAD) && (x >= RAD+1) && (x < Ww-RAD);
    nms[i] = (mm[i] != 0.0f && border) ? s[i] : 0.0f;
}

// ---------------- exact top-K via bit-pattern histogram + bitonic sort ----------------
__global__ void k_zero_u32(unsigned* __restrict__ p, int n){
    int i = blockIdx.x * blockDim.x + threadIdx.x;
    if (i < n) p[i] = 0u;
}

__global__ void k_hist(const float* __restrict__ nms, unsigned* __restrict__ hist){
    int i = blockIdx.x * blockDim.x + threadIdx.x;
    if (i >= Bn*HW) return;
    int b = i / HW;
    unsigned key = __float_as_uint(nms[i]);   // nms >= 0 -> bits are order-preserving
    atomicAdd(&hist[(unsigned)b*65536u + (key >> 16)], 1u);
}

__global__ void k_thresh(const unsigned* __restrict__ hist, unsigned* __restrict__ ctr){
    int b = threadIdx.x;
    if (b >= Bn) return;
    unsigned cum = 0, above = 0; int T = 0;
    for (int bin = 65535; bin >= 0; --bin){
        unsigned c = hist[(unsigned)b*65536u + bin];
        if (cum + c >= (unsigned)Kk){ T = bin; above = cum; break; }
        cum += c;
    }
    ctr[b*8 + 2] = (unsigned)T;
    ctr[b*8 + 3] = above;      // exact count of elements strictly above bin T
}

__global__ void k_compact(const float* __restrict__ nms, unsigned* __restrict__ ctr,
                          unsigned* __restrict__ keys, unsigned* __restrict__ idxs,
                          unsigned* __restrict__ eqK, unsigned* __restrict__ eqI){
    int i = blockIdx.x * blockDim.x + threadIdx.x;
    if (i >= Bn*HW) return;
    int b = i / HW; unsigned li = (unsigned)(i - b*HW);
    unsigned key = __float_as_uint(nms[i]);
    unsigned bin = key >> 16;
    unsigned T = ctr[b*8 + 2];
    if (bin > T){
        unsigned p = atomicAdd(&ctr[b*8 + 0], 1u);
        if (p < (unsigned)Kk){ keys[b*Kk + p] = key; idxs[b*Kk + p] = li; }
    } else if (bin == T){
        unsigned p = atomicAdd(&ctr[b*8 + 1], 1u);
        if (p < 4096u){ eqK[b*4096 + p] = key; eqI[b*4096 + p] = li; }
    }
}

__global__ void k_fill(const unsigned* __restrict__ ctr, unsigned* __restrict__ keys, unsigned* __restrict__ idxs,
                       const unsigned* __restrict__ eqK, const unsigned* __restrict__ eqI){
    int t = blockIdx.x * blockDim.x + threadIdx.x;
    if (t >= Bn*Kk) return;
    int b = t / Kk; unsigned p = (unsigned)(t - b*Kk);
    unsigned above = ctr[b*8 + 3];
    unsigned eqn   = ctr[b*8 + 1];
    if (p >= above){
        unsigned j = p - above;
        if (j < eqn && j < 4096u){ keys[b*Kk + p] = eqK[b*4096 + j]; idxs[b*Kk + p] = eqI[b*4096 + j]; }
        else                     { keys[b*Kk + p] = 0u;              idxs[b*Kk + p] = 0u; }
    }
}

// One block per batch: bitonic sort of 2048 (key,idx) pairs in LDS.
// Sort ascending on ~(key<<32 | (~idx)) == descending key, ascending idx (matches lax.top_k).
__global__ void k_sort(unsigned* __restrict__ keys, unsigned* __restrict__ idxs){
    __shared__ unsigned long long s[Kk];
    int b = blockIdx.x;
    for (int p = threadIdx.x; p < Kk; p += blockDim.x){
        unsigned long long comb = ((unsigned long long)keys[b*Kk + p] << 32) |
                                  (unsigned long long)(0xFFFFFFFFu - idxs[b*Kk + p]);
        s[p] = ~comb;
    }
    __syncthreads();
    for (int kk2 = 2; kk2 <= Kk; kk2 <<= 1){
        for (int j = kk2 >> 1; j > 0; j >>= 1){
            for (int p = threadIdx.x; p < Kk; p += blockDim.x){
                int ixj = p ^ j;
                if (ixj > p){
                    bool up = ((p & kk2) == 0);
                    unsigned long long a = s[p], c = s[ixj];
                    if ((a > c) == up){ s[p] = c; s[ixj] = a; }
                }
            }
            __syncthreads();
        }
    }
    for (int p = threadIdx.x; p < Kk; p += blockDim.x){
        unsigned long long comb = ~s[p];
        keys[b*Kk + p] = (unsigned)(comb >> 32);
        idxs[b*Kk + p] = 0xFFFFFFFFu - (unsigned)(comb & 0xFFFFFFFFu);
    }
}

// ---------------- soft-argmax refine via V_WMMA_F32_16X16X4_F32 ----------------
// D(16kp x 16) = sum_k A(16kp x 4) * B(4 x 16); B columns = [1, gx, gy, gx^2, gy^2, 0...]
__global__ void k_refine(const float* __restrict__ scores, const unsigned* __restrict__ idxs,
                         float* __restrict__ outKxy, float* __restrict__ outScore,
                         float* __restrict__ outDisp, float* __restrict__ samp){
    __shared__ float Axp[128][28];   // exp-weights, padded K=25 -> 28
    __shared__ float Btab[16][28];   // constant moment basis, [column n][k]
    __shared__ float Mom[128][5];    // ssum, Sx, Sy, Sxx, Syy
    __shared__ int   Kxi[128], Kyi[128];

    int blk   = blockIdx.x;              // Bn * (Kk/128) blocks
    int b     = blk / (Kk/128);
    int chunk = blk % (Kk/128);
    int t     = threadIdx.x;             // 256 threads = 8 waves x 16 keypoints

    // Branch-free fill of the constant B table (once per block, outside hot loop)
    for (int idx = t; idx < 16*28; idx += 256){
        int n = idx / 28, k = idx - n*28;
        float gx = (float)(k % 5 - 2);
        float gy = (float)(k / 5 - 2);
        float v = 0.0f;
        v = (n == 0) ? 1.0f  : v;
        v = (n == 1) ? gx    : v;
        v = (n == 2) ? gy    : v;
        v = (n == 3) ? gx*gx : v;
        v = (n == 4) ? gy*gy : v;
        Btab[n][k] = (k < 25) ? v : 0.0f;
    }

    if (t < 128){
        int kp = chunk*128 + t;
        unsigned idx = idxs[b*Kk + kp];
        int ky = (int)(idx / Ww), kx = (int)(idx % Ww);
        Kxi[t] = kx; Kyi[t] = ky;
        float vals[25]; float mx = -INFINITY;
        #pragma unroll
        for (int p = 0; p < 25; ++p){
            int y = ky + p/5 - RAD, x = kx + p%5 - RAD;
            float v = (y >= 0 && y < Hh && x >= 0 && x < Ww) ? scores[(b*Hh + y)*Ww + x] : 0.0f;
            vals[p] = v; mx = fmaxf(mx, v);
        }
        #pragma unroll
        for (int p = 0; p < 25; ++p) Axp[t][p] = __expf((vals[p] - mx) * 10.0f);  // 1/TEMP
        Axp[t][25] = 0.0f; Axp[t][26] = 0.0f; Axp[t][27] = 0.0f;
    }
    __syncthreads();

    // WMMA moment GEMM: all 32 lanes of every wave active (EXEC all-1s).
    // Loop body is just ds_load_b64 (A) + ds_load_b64 (B) + v_wmma.
    int lane = t & 31, wv = t >> 5;
    int m = lane & 15, half = lane >> 4;
    int kpB = wv * 16;
    v8f acc = {};
    #pragma unroll
    for (int j = 0; j < 7; ++j){
        int ka = 4*j + 2*half;                 // lanes 0-15: K={0,1}; lanes 16-31: K={2,3}
        v2f a, bb;
        a.x  = Axp[kpB + m][ka];
        a.y  = Axp[kpB + m][ka + 1];
        bb.x = Btab[m][ka];
        bb.y = Btab[m][ka + 1];
        acc = __builtin_amdgcn_wmma_f32_16x16x4_f32(false, a, false, bb, (short)0, acc, false, false);
    }
    // D layout: lane<16 -> (M=v, N=lane); lane>=16 -> (M=v+8, N=lane-16)
    if (m < 5){
        int baseRow = kpB + (half ? 8 : 0);
        #pragma unroll
        for (int v = 0; v < 8; ++v) Mom[baseRow + v][m] = acc[v];
    }
    __syncthreads();

    if (t < 128){
        float ssum = Mom[t][0], Sx = Mom[t][1], Sy = Mom[t][2], Sxx = Mom[t][3], Syy = Mom[t][4];
        float inv = 1.0f / ssum;                         // ssum >= 1 (max term is exp(0))
        float X = Sx*inv, Y = Sy*inv;
        float disp = (Sxx*inv - X*X + Syy*inv - Y*Y) * 0.25f;   // / r^2, r=2
        int kp = chunk*128 + t;
        float nx = ((float)Kxi[t] + X) / (float)(Ww-1) * 2.0f - 1.0f;
        float ny = ((float)Kyi[t] + Y) / (float)(Hh-1) * 2.0f - 1.0f;
        size_t g = (size_t)b*Kk + kp;
        outKxy[g*2 + 0] = nx;
        outKxy[g*2 + 1] = ny;
        outDisp[g] = disp;
        float xp = (nx + 1.0f) * 0.5f * (float)(Ww-1);
        float yp = (ny + 1.0f) * 0.5f * (float)(Hh-1);
        samp[g*2 + 0] = xp; samp[g*2 + 1] = yp;
        // bilinear score sample (c==1 channel)
        float x0 = floorf(xp), y0 = floorf(yp);
        float wx = xp - x0, wy = yp - y0;
        int x0i = clampi((int)x0, 0, Ww-1), x1i = clampi((int)x0 + 1, 0, Ww-1);
        int y0i = clampi((int)y0, 0, Hh-1), y1i = clampi((int)y0 + 1, 0, Hh-1);
        const float* sb = scores + (size_t)b*HW;
        float v00 = sb[y0i*Ww + x0i], v01 = sb[y0i*Ww + x1i];
        float v10 = sb[y1i*Ww + x0i], v11 = sb[y1i*Ww + x1i];
        outScore[g] = v00*(1-wx)*(1-wy) + v01*wx*(1-wy) + v10*(1-wx)*wy + v11*wx*wy;
    }
}

// ---------------- descriptor bilinear gather + L2 normalize ----------------
__global__ void k_desc(const float* __restrict__ dmap, const float* __restrict__ samp,
                       float* __restrict__ outDesc){
    __shared__ float red[4];
    int gk = blockIdx.x;                 // Bn*Kk blocks, one keypoint each
    int b  = gk / Kk;
    int c  = threadIdx.x;                // 128 threads = 128 channels
    float xp = samp[(size_t)gk*2 + 0], yp = samp[(size_t)gk*2 + 1];
    float x0 = floorf(xp), y0 = floorf(yp);
    float wx = xp - x0, wy = yp - y0;
    int x0i = clampi((int)x0, 0, Ww-1), x1i = clampi((int)x0 + 1, 0, Ww-1);
    int y0i = clampi((int)y0, 0, Hh-1), y1i = clampi((int)y0 + 1, 0, Hh-1);
    const float* base = dmap + (size_t)(b*Cc + c) * HW;
    __builtin_prefetch(base + y1i*Ww + x0i, 0, 1);      // prefetch second bilinear row
    float v00 = base[y0i*Ww + x0i], v01 = base[y0i*Ww + x1i];
    float v10 = base[y1i*Ww + x0i], v11 = base[y1i*Ww + x1i];
    float acc = v00*(1-wx)*(1-wy) + v01*wx*(1-wy) + v10*(1-wx)*wy + v11*wx*wy;
    // wave32 + LDS norm reduction
    float sq = acc * acc;
    for (int off = 16; off > 0; off >>= 1) sq += __shfl_down(sq, off, 32);
    if ((threadIdx.x & 31) == 0) red[threadIdx.x >> 5] = sq;
    __syncthreads();
    float tot = red[0] + red[1] + red[2] + red[3];
    float nrm = fmaxf(sqrtf(tot), 1e-12f);
    outDesc[(size_t)gk*Cc + c] = acc / nrm;
}

// ---------------- orchestration ----------------
extern "C" void kernel_launch(void* const* d_in, const int* in_sizes, int n_in,
                              void* d_out, int out_size, void* d_ws, size_t ws_size,
                              hipStream_t stream){
    const float* scores = (const float*)d_in[0];   // (2,1,512,512)
    const float* dmap   = (const float*)d_in[1];   // (2,128,512,512)

    const size_t BHW = (size_t)Bn*HW;
    float* f = (float*)d_ws;
    float* tmpB   = f;
    float* mm     = f + 1*BHW;
    float* mmP    = f + 2*BHW;
    float* ss     = f + 3*BHW;
    float* nms    = f + 4*BHW;
    float* samp   = f + 5*BHW;                        // Bn*Kk*2
    unsigned* hist = (unsigned*)(samp + (size_t)Bn*Kk*2);
    unsigned* ctr  = hist + (size_t)Bn*65536;         // 8 words per batch
    unsigned* keys = ctr  + Bn*8;
    unsigned* idxs = keys + Bn*Kk;
    unsigned* eqK  = idxs + Bn*Kk;
    unsigned* eqI  = eqK  + (size_t)Bn*4096;
    size_t need = ((char*)(eqI + (size_t)Bn*4096)) - (char*)d_ws;
    if (ws_size < need) return;

    dim3 blk(256);
    dim3 grdEl((unsigned)((BHW + 255) / 256));

    // _simple_nms (5 max-pool passes, column pass fused with elementwise logic)
    k_mp_rows<<<grdEl, blk, 0, stream>>>(scores, tmpB);
    k_cols_eq<<<grdEl, blk, 0, stream>>>(tmpB, scores, mm);
    for (int it = 0; it < 2; ++it){
        k_mp_rows<<<grdEl, blk, 0, stream>>>(mm, tmpB);
        k_cols_suppress<<<grdEl, blk, 0, stream>>>(tmpB, scores, mmP, ss);
        k_mp_rows<<<grdEl, blk, 0, stream>>>(ss, tmpB);
        k_cols_combine<<<grdEl, blk, 0, stream>>>(tmpB, ss, mmP, mm);
    }
    k_nms_border<<<grdEl, blk, 0, stream>>>(scores, mm, nms);

    // exact top-2048 per batch
    int zn = Bn*65536 + Bn*8;
    k_zero_u32<<<(zn + 255)/256, blk, 0, stream>>>(hist, zn);
    k_hist<<<grdEl, blk, 0, stream>>>(nms, hist);
    k_thresh<<<1, 32, 0, stream>>>(hist, ctr);
    k_compact<<<grdEl, blk, 0, stream>>>(nms, ctr, keys, idxs, eqK, eqI);
    k_fill<<<(Bn*Kk + 255)/256, blk, 0, stream>>>(ctr, keys, idxs, eqK, eqI);
    k_sort<<<Bn, 256, 0, stream>>>(keys, idxs);

    // outputs: (kxy, desc, kptscores, disp) concatenated flat
    float* outKxy   = (float*)d_out;
    float* outDesc  = outKxy  + (size_t)Bn*Kk*2;
    float* outScore = outDesc + (size_t)Bn*Kk*Cc;
    float* outDisp  = outScore + (size_t)Bn*Kk;

    k_refine<<<Bn*(Kk/128), 256, 0, stream>>>(scores, idxs, outKxy, outScore, outDisp, samp);
    k_desc<<<Bn*Kk, Cc, 0, stream>>>(dmap, samp, outDesc);
}